// Capsule_16484084482446
// MI455X (gfx1250) — compile-verified
//
#include <hip/hip_runtime.h>

// Capsule routing on MI455X (gfx1250).
// Exact simplification: softmax over the capsule axis makes the routing
// coefficients sum to 1 per (b,n), so outputs[b,j,d] == (sum_n x[b,n,:]) @ W.
// Stage 1: bandwidth-bound column reduction of x (64 MiB read, ~2.8us @ 23.3TB/s).
// Stage 2: 32x160x128 f32 GEMM via V_WMMA_F32_16X16X4_F32.

#define NCAP   10
#define DCAP   16
#define NOUT   (NCAP * DCAP)   // 160
#define DIN    128
#define NROWS  4096
#define BATCH  32
#define NCHUNK 8
#define ROWS_PER_CHUNK (NROWS / NCHUNK)     // 512
#define ROWS_PER_WAVE  (ROWS_PER_CHUNK / 8) // 64

typedef __attribute__((ext_vector_type(2))) float v2f;
typedef __attribute__((ext_vector_type(8))) float v8f;

// ---------------------------------------------------------------------------
// Stage 1: partials[b][chunk][k] = sum over 512 rows of x[b][n][k].
// Block = 8 wave32s; lane owns a float4 (4 consecutive k), wave owns 64 rows.
// ---------------------------------------------------------------------------
__global__ void capsule_colsum_kernel(const float* __restrict__ x,
                                      float* __restrict__ partials) {
    const int b     = blockIdx.x;          // 0..31
    const int chunk = blockIdx.y;          // 0..7
    const int wave  = threadIdx.x >> 5;    // 0..7
    const int lane  = threadIdx.x & 31;

    const float* xrow = x + ((size_t)b * NROWS +
                             (size_t)chunk * ROWS_PER_CHUNK +
                             (size_t)wave * ROWS_PER_WAVE) * DIN;

    float4 acc = make_float4(0.f, 0.f, 0.f, 0.f);
#pragma unroll 4
    for (int r = 0; r < ROWS_PER_WAVE; ++r) {
        const float4 v = *(const float4*)(xrow + (size_t)r * DIN + lane * 4);
        acc.x += v.x; acc.y += v.y; acc.z += v.z; acc.w += v.w;
    }

    __shared__ float lds[8 * DIN];         // 4 KB
    *(float4*)(lds + wave * DIN + lane * 4) = acc;
    __syncthreads();

    const int t = threadIdx.x;
    if (t < DIN) {
        float s = 0.f;
#pragma unroll
        for (int w = 0; w < 8; ++w) s += lds[w * DIN + t];
        partials[((size_t)b * NCHUNK + chunk) * DIN + t] = s;
    }
}

// ---------------------------------------------------------------------------
// Stage 2: out[b][col] = sum_k xsum[b][k] * W[k][col], via f32 WMMA.
// One wave per 16x16 output tile; grid = (2 row tiles, 10 col tiles).
// A fragment (16x4 f32): lane&15 = M, (lane>>4)*2 + vgpr = K.
// B fragment (4x16 f32): lane&15 = N, (lane>>4)*2 + vgpr = K.
// C/D (16x16 f32): vgpr r -> M = r + (lane>>4)*8, N = lane&15.
// ---------------------------------------------------------------------------
__global__ void capsule_wmma_kernel(const float* __restrict__ partials,
                                    const float* __restrict__ W,
                                    float* __restrict__ out) {
    const int mi   = blockIdx.x;           // 0..1
    const int nj   = blockIdx.y;           // 0..9
    const int lane = threadIdx.x;          // 0..31

    __shared__ float asum[16][DIN];        // 8 KB
    for (int idx = lane; idx < 16 * DIN; idx += 32) {
        const int bl = idx >> 7;
        const int k  = idx & (DIN - 1);
        const int b  = mi * 16 + bl;
        float s = 0.f;
#pragma unroll
        for (int c = 0; c < NCHUNK; ++c)
            s += partials[((size_t)b * NCHUNK + c) * DIN + k];
        asum[bl][k] = s;
    }
    __syncthreads();

    const int half = lane >> 4;            // lane half selects K pair
    const int l15  = lane & 15;
    const int col  = nj * 16 + l15;

    v8f acc = {};
#pragma unroll
    for (int kk = 0; kk < DIN / 4; ++kk) {
        const int kb = kk * 4 + half * 2;
        v2f a, bf;
        a.x  = asum[l15][kb];
        a.y  = asum[l15][kb + 1];
        bf.x = W[(size_t)kb * NOUT + col];
        bf.y = W[((size_t)kb + 1) * NOUT + col];
        acc = __builtin_amdgcn_wmma_f32_16x16x4_f32(
            /*neg_a=*/false, a, /*neg_b=*/false, bf,
            /*c_mod=*/(short)0, acc, /*reuse_a=*/false, /*reuse_b=*/false);
    }

#pragma unroll
    for (int r = 0; r < 8; ++r) {
        const int m = r + half * 8;
        out[((size_t)(mi * 16 + m)) * NOUT + col] = acc[r];
    }
}

// ---------------------------------------------------------------------------
extern "C" void kernel_launch(void* const* d_in, const int* in_sizes, int n_in,
                              void* d_out, int out_size, void* d_ws, size_t ws_size,
                              hipStream_t stream) {
    const float* x = (const float*)d_in[0];   // (32, 4096, 128) f32
    const float* W = (const float*)d_in[1];   // (128, 160)      f32
    float* out      = (float*)d_out;          // (32, 10, 16)    f32
    float* partials = (float*)d_ws;           // 32*8*128 floats = 128 KB

    dim3 g1(BATCH, NCHUNK);
    capsule_colsum_kernel<<<g1, 256, 0, stream>>>(x, partials);

    dim3 g2(2, NCAP);
    capsule_wmma_kernel<<<g2, 32, 0, stream>>>(partials, W, out);
}